// FVGAE_82042465288961
// MI455X (gfx1250) — compile-verified
//
#include <hip/hip_runtime.h>
#include <stdint.h>

#define FD 256            // feature dim
#define CHUNK 32          // K-rows of W staged into LDS per TDM transfer
#define LRELU_ALPHA 0.2f

typedef float v2f __attribute__((ext_vector_type(2)));
typedef float v8f __attribute__((ext_vector_type(8)));
typedef unsigned int u32x4 __attribute__((ext_vector_type(4)));
typedef int i32x4 __attribute__((ext_vector_type(4)));
typedef int i32x8 __attribute__((ext_vector_type(8)));

#if defined(__has_builtin)
#if __has_builtin(__builtin_amdgcn_tensor_load_to_lds)
#define USE_TDM 1
#endif
#endif
#ifndef USE_TDM
#define USE_TDM 0
#endif

// ---------------------------------------------------------------------------
// Stage one CHUNKxFD f32 tile of W into LDS.
// TDM path: one wave issues a tensor_load_to_lds with a 2-D D# descriptor
//   (group0: count=1, lds_addr, global_addr, type=2;
//    group1: data_size=4B, tensor_dim0=FD, tensor_dim1=CHUNK,
//            tile_dim0=FD, tile_dim1=CHUNK, tensor_dim0_stride=FD).
// Fallback: cooperative float4 copy by all 128 threads.
// ---------------------------------------------------------------------------
__device__ __forceinline__ void stage_issue(const float* __restrict__ gsrc,
                                            float* __restrict__ ldst,
                                            int wave, unsigned tid)
{
#if USE_TDM
    if (wave == 0) {
        const unsigned long long ga = (unsigned long long)(uintptr_t)gsrc;
        u32x4 g0;
        g0.x = 1u;                                   // count=1 (valid descriptor)
        g0.y = (unsigned)(uintptr_t)ldst;            // LDS byte address
        g0.z = (unsigned)ga;                         // global_addr[31:0]
        g0.w = (unsigned)(ga >> 32) | (2u << 30);    // global_addr[56:32] | type=2
        // group1: [17:16]=data_size(4B)=2 ; dim0=256 @[79:48] ; dim1=32 @[111:80]
        //         tile_dim0=256 @[127:112] ; tile_dim1=32 @[143:128] ; stride0=256 @[207:160]
        i32x8 g1 = { 0x00020000, 0x01000000, 0x00200000, 0x01000000,
                     CHUNK, FD, 0, 0 };
        i32x4 z4 = { 0, 0, 0, 0 };
#if __clang_major__ >= 23
        i32x8 z8 = { 0, 0, 0, 0, 0, 0, 0, 0 };
        __builtin_amdgcn_tensor_load_to_lds(g0, g1, z4, z4, z8, 0);
#else
        __builtin_amdgcn_tensor_load_to_lds(g0, g1, z4, z4, 0);
#endif
    }
#else
    for (unsigned i = tid; i < CHUNK * FD / 4; i += 128u)
        ((float4*)ldst)[i] = ((const float4*)gsrc)[i];
#endif
}

__device__ __forceinline__ void stage_fence(int wave)
{
#if USE_TDM
    if (wave == 0) __builtin_amdgcn_s_wait_tensorcnt(0);  // TENSORcnt==0 -> LDS valid
#endif
    __syncthreads();
}

// ---------------------------------------------------------------------------
// GEMM: C[M,256] = act( [A1 | A2][M,K] @ W[K,256] (+ bias) )
//   K==256: A entirely from A1.  K==512: cols 0..255 from A1, rest from A2.
//   act: 0 = identity, 1 = relu.
// 128-thread block (4 wave32 waves) computes a 32-row x 256-col strip.
// W is streamed through LDS in CHUNK-row tiles via the Tensor Data Mover,
// double-buffered against V_WMMA_F32_16X16X4_F32 compute.
// Each wave: 2 A fragments (rows r*16) x 4 N-tiles = 8 v8f accumulators;
// each LDS B fragment feeds two WMMAs.
// ---------------------------------------------------------------------------
__global__ __launch_bounds__(128)
void gemm_f32_wmma(const float* __restrict__ A1, const float* __restrict__ A2,
                   const float* __restrict__ W,  const float* __restrict__ bias,
                   float* __restrict__ C, int M, int K, int act)
{
    __shared__ float wbuf[2][CHUNK * FD];   // 2 x 32KB double buffer

    const int lane = threadIdx.x & 31;
    const int wave = threadIdx.x >> 5;      // 0..3
    const int l    = lane & 15;
    const int hi   = lane >> 4;             // 0 or 1
    const int m0   = blockIdx.x * 32;
    if (m0 >= M) return;                    // uniform per block

    const int nchunk = K / CHUNK;

    v8f acc[2][4];
#pragma unroll
    for (int r = 0; r < 2; ++r)
#pragma unroll
        for (int t = 0; t < 4; ++t)
#pragma unroll
            for (int j = 0; j < 8; ++j) acc[r][t][j] = 0.0f;

    const size_t arow0 = (size_t)(m0 + l) * FD;
    const size_t arow1 = (size_t)(m0 + 16 + l) * FD;

    stage_issue(W, wbuf[0], wave, threadIdx.x);
    stage_fence(wave);

    for (int c = 0; c < nchunk; ++c) {
        if (c + 1 < nchunk)
            stage_issue(W + (size_t)(c + 1) * CHUNK * FD,
                        wbuf[(c + 1) & 1], wave, threadIdx.x);

        const float* Ap    = (c * CHUNK < FD) ? A1 : A2;
        const int    kbase = (c * CHUNK) & (FD - 1);
        const float* Bl    = wbuf[c & 1];

        if (c + 1 < nchunk) {                 // global_prefetch_b8 one chunk ahead
            const float* ApN = ((c + 1) * CHUNK < FD) ? A1 : A2;
            const int    kbN = ((c + 1) * CHUNK) & (FD - 1);
            __builtin_prefetch(ApN + arow0 + kbN, 0, 1);
            __builtin_prefetch(ApN + arow1 + kbN, 0, 1);
        }

#pragma unroll
        for (int kk = 0; kk < CHUNK; kk += 4) {
            const int ka = kbase + kk + hi * 2;        // even -> 8B aligned
            const v2f a0 = *(const v2f*)(Ap + arow0 + ka);
            const v2f a1 = *(const v2f*)(Ap + arow1 + ka);
#pragma unroll
            for (int t = 0; t < 4; ++t) {
                const int n = wave * 64 + t * 16 + l;
                v2f b;
                b.x = Bl[(kk + hi * 2) * FD + n];
                b.y = Bl[(kk + hi * 2 + 1) * FD + n];
                acc[0][t] = __builtin_amdgcn_wmma_f32_16x16x4_f32(
                                false, a0, false, b, (short)0, acc[0][t], false, false);
                acc[1][t] = __builtin_amdgcn_wmma_f32_16x16x4_f32(
                                false, a1, false, b, (short)0, acc[1][t], false, false);
            }
        }
        stage_fence(wave);   // next buffer complete before it is consumed
    }

#pragma unroll
    for (int r = 0; r < 2; ++r)
#pragma unroll
        for (int t = 0; t < 4; ++t) {
            const int col = wave * 64 + t * 16 + l;
            const float bv = bias ? bias[col] : 0.0f;
#pragma unroll
            for (int j = 0; j < 8; ++j) {
                const int row = m0 + r * 16 + hi * 8 + j;
                float v = acc[r][t][j] + bv;
                if (act == 1) v = fmaxf(v, 0.0f);
                C[(size_t)row * FD + col] = v;
            }
        }
}

// ---------------------------------------------------------------------------
// SpMM scatter:  Y[dst[e], :] += w[e] * X[src[e], :]
// 4 edges per 256-thread block; 64 lanes x float4 per edge (B128 gathers,
// fp32 atomics resolve in the 192MB L2 since Y is ~102MB).  Y pre-zeroed.
// ---------------------------------------------------------------------------
__global__ __launch_bounds__(256)
void spmm_kernel(const float* __restrict__ X, const int* __restrict__ src,
                 const int* __restrict__ dst, const float* __restrict__ w,
                 float* __restrict__ Y, int E)
{
    const int e = blockIdx.x * 4 + (threadIdx.x >> 6);
    if (e >= E) return;
    const int c = (threadIdx.x & 63) * 4;
    const int s = src[e];
    const int d = dst[e];
    const float we = w[e];
    const float4 x = *(const float4*)(X + (size_t)s * FD + c);
    float* yp = Y + (size_t)d * FD + c;
    atomicAdd(yp + 0, we * x.x);
    atomicAdd(yp + 1, we * x.y);
    atomicAdd(yp + 2, we * x.z);
    atomicAdd(yp + 3, we * x.w);
}

// ---------------------------------------------------------------------------
// Y = leaky_relu(Y + bias)  (vectorized, in place).
// ---------------------------------------------------------------------------
__global__ __launch_bounds__(256)
void bias_leaky(float* __restrict__ Y, const float* __restrict__ b, long long n4)
{
    const long long i = (long long)blockIdx.x * blockDim.x + threadIdx.x;
    if (i >= n4) return;
    const int cb = (int)(i & (FD / 4 - 1)) * 4;
    float4 v  = ((float4*)Y)[i];
    const float4 bb = *(const float4*)(b + cb);
    v.x += bb.x; v.y += bb.y; v.z += bb.z; v.w += bb.w;
    v.x = v.x > 0.f ? v.x : LRELU_ALPHA * v.x;
    v.y = v.y > 0.f ? v.y : LRELU_ALPHA * v.y;
    v.z = v.z > 0.f ? v.z : LRELU_ALPHA * v.z;
    v.w = v.w > 0.f ? v.w : LRELU_ALPHA * v.w;
    ((float4*)Y)[i] = v;
}

// ---------------------------------------------------------------------------
// KL(N(mu,s1) || N(0,s2)) elementwise sum; s2 = sigma(1.0) constant.
// ---------------------------------------------------------------------------
__global__ __launch_bounds__(256)
void kld_reduce(const float* __restrict__ mu, const float* __restrict__ ls,
                long long n, float* __restrict__ acc)
{
    const float ls2  = 0.1f + 0.9f * log1pf(expf(1.0f));
    const float s2   = expf(ls2);
    const float inv2 = 1.0f / (2.0f * s2 * s2);

    float part = 0.0f;
    for (long long i = (long long)blockIdx.x * blockDim.x + threadIdx.x;
         i < n; i += (long long)gridDim.x * blockDim.x) {
        const float m  = mu[i];
        const float l1 = 0.1f + 0.9f * log1pf(expf(fminf(ls[i], 0.4f)));
        const float s1 = expf(l1);
        part += (ls2 - l1) + (s1 * s1 + m * m) * inv2 - 0.5f;
    }

    __shared__ float red[256];
    red[threadIdx.x] = part;
    __syncthreads();
    for (int s = 128; s > 0; s >>= 1) {
        if (threadIdx.x < (unsigned)s) red[threadIdx.x] += red[threadIdx.x + s];
        __syncthreads();
    }
    if (threadIdx.x == 0) atomicAdd(acc, red[0]);
}

__global__ void kld_finalize(const float* __restrict__ acc, float* __restrict__ out,
                             float invNU, float invNI)
{
    if (threadIdx.x == 0)
        out[0] = 0.5f * acc[0] * invNU + acc[1] * invNI;   // BETA = 0.5
}

// ---------------------------------------------------------------------------
// Host-side orchestration
// ---------------------------------------------------------------------------
extern "C" void kernel_launch(void* const* d_in, const int* in_sizes, int n_in,
                              void* d_out, int out_size, void* d_ws, size_t ws_size,
                              hipStream_t stream)
{
    const int NU = 100000, NI = 100000, E = 1600000;
    const size_t NR = (size_t)NU * FD;

    const float* ufea = (const float*)d_in[0];
    const float* vfea = (const float*)d_in[1];
    const int*   eu   = (const int*)  d_in[2];
    const int*   ev   = (const int*)  d_in[3];
    const float* wuv  = (const float*)d_in[4];
    const float* wvu  = (const float*)d_in[5];
    const float* dW1 = (const float*)d_in[6];  const float* db1 = (const float*)d_in[7];
    const float* dW2 = (const float*)d_in[8];  const float* db2 = (const float*)d_in[9];
    const float* dW3 = (const float*)d_in[10]; const float* db3 = (const float*)d_in[11];
    const float* dW4 = (const float*)d_in[12]; const float* db4 = (const float*)d_in[13];
    const float* dWu = (const float*)d_in[14]; const float* dbu = (const float*)d_in[15];
    const float* dWi = (const float*)d_in[16]; const float* dbi = (const float*)d_in[17];
    const float* lW1 = (const float*)d_in[18]; const float* lb1 = (const float*)d_in[19];
    const float* lW2 = (const float*)d_in[20]; const float* lb2 = (const float*)d_in[21];
    const float* W3m = (const float*)d_in[22]; const float* b3m = (const float*)d_in[23];
    const float* W3s = (const float*)d_in[24]; const float* b3s = (const float*)d_in[25];
    const float* W4m = (const float*)d_in[26]; const float* b4m = (const float*)d_in[27];
    const float* W4s = (const float*)d_in[28]; const float* b4s = (const float*)d_in[29];
    const float* Wum = (const float*)d_in[30]; const float* bum = (const float*)d_in[31];
    const float* Wus = (const float*)d_in[32]; const float* bus = (const float*)d_in[33];
    const float* Wim = (const float*)d_in[34]; const float* bim = (const float*)d_in[35];
    const float* Wis = (const float*)d_in[36]; const float* bis = (const float*)d_in[37];

    float* out    = (float*)d_out;
    float* o_user = out;
    float* o_item = out + NR;
    float* o_mu   = out + 2 * NR;
    float* o_mi   = out + 3 * NR;
    float* o_lu   = out + 4 * NR;
    float* o_li   = out + 5 * NR;
    float* o_kld  = out + 6 * NR;

    float* ws = (float*)d_ws;
    float* t0 = ws;
    float* t1 = ws + NR;
    float* t2 = ws + 2 * NR;
    float* U  = ws + 3 * NR;
    float* V  = ws + 4 * NR;
    float* acc = ws + 5 * NR;

    auto gemm = [&](const float* A1, const float* A2, const float* Wm,
                    const float* b, float* Co, int M, int K, int act) {
        gemm_f32_wmma<<<M / 32, 128, 0, stream>>>(A1, A2 ? A2 : A1, Wm, b, Co, M, K, act);
    };
    // spmm_uv: Y[eu] += wuv * X[ev]   /   spmm_vu: Y[ev] += wvu * X[eu]
    auto spmm = [&](const float* X, const int* s, const int* d, const float* w,
                    float* Y, int Mout) {
        hipMemsetAsync(Y, 0, (size_t)Mout * FD * sizeof(float), stream);
        spmm_kernel<<<(E + 3) / 4, 256, 0, stream>>>(X, s, d, w, Y, E);
    };
    auto bleaky = [&](float* Y, const float* b, int M) {
        const long long n4 = (long long)M * (FD / 4);
        bias_leaky<<<(unsigned)((n4 + 255) / 256), 256, 0, stream>>>(Y, b, n4);
    };

    // ================= dgcn layer =================
    gemm(ufea, nullptr, dW1, nullptr, t0, NU, 256, 0);
    spmm(t0, eu, ev, wvu, t1, NI);  bleaky(t1, db1, NI);
    gemm(t1, nullptr, dW3, nullptr, t0, NI, 256, 0);
    spmm(t0, ev, eu, wuv, t2, NU);  bleaky(t2, db3, NU);           // uh
    gemm(t2, ufea, dWu, dbu, U, NU, 512, 1);                       // u_new
    gemm(vfea, nullptr, dW2, nullptr, t0, NI, 256, 0);
    spmm(t0, ev, eu, wuv, t1, NU);  bleaky(t1, db2, NU);
    gemm(t1, nullptr, dW4, nullptr, t0, NU, 256, 0);
    spmm(t0, eu, ev, wvu, t2, NI);  bleaky(t2, db4, NI);           // vh
    gemm(t2, vfea, dWi, dbi, V, NI, 512, 1);                       // v_new

    // ================= last layer: item branch =================
    gemm(V, nullptr, lW2, nullptr, t0, NI, 256, 0);
    spmm(t0, ev, eu, wuv, t1, NU);  bleaky(t1, lb2, NU);           // ih [NU]
    gemm(t1, nullptr, W4m, nullptr, t0, NU, 256, 0);
    spmm(t0, eu, ev, wvu, t2, NI);  bleaky(t2, b4m, NI);
    gemm(t2, V, Wim, bim, o_mi, NI, 512, 0);                       // mean_i
    gemm(t1, nullptr, W4s, nullptr, t0, NU, 256, 0);
    spmm(t0, eu, ev, wvu, t2, NI);  bleaky(t2, b4s, NI);
    gemm(t2, V, Wis, bis, o_li, NI, 512, 0);                       // logstd_i

    // ================= last layer: user branch =================
    gemm(U, nullptr, lW1, nullptr, t0, NU, 256, 0);
    spmm(t0, eu, ev, wvu, t1, NI);  bleaky(t1, lb1, NI);           // uh [NI]
    gemm(t1, nullptr, W3m, nullptr, t0, NI, 256, 0);
    spmm(t0, ev, eu, wuv, t2, NU);  bleaky(t2, b3m, NU);
    gemm(t2, U, Wum, bum, o_mu, NU, 512, 0);                       // mean_u
    gemm(t1, nullptr, W3s, nullptr, t0, NI, 256, 0);
    spmm(t0, ev, eu, wuv, t2, NU);  bleaky(t2, b3s, NU);
    gemm(t2, U, Wus, bus, o_lu, NU, 512, 0);                       // logstd_u

    hipMemcpyAsync(o_user, o_mu, NR * sizeof(float), hipMemcpyDeviceToDevice, stream);
    hipMemcpyAsync(o_item, o_mi, NR * sizeof(float), hipMemcpyDeviceToDevice, stream);

    hipMemsetAsync(acc, 0, 2 * sizeof(float), stream);
    kld_reduce<<<1024, 256, 0, stream>>>(o_mu, o_lu, (long long)NR, acc + 0);
    kld_reduce<<<1024, 256, 0, stream>>>(o_mi, o_li, (long long)NR, acc + 1);
    kld_finalize<<<1, 1, 0, stream>>>(acc, o_kld, 1.0f / NU, 1.0f / NI);
}